// CustomGeneralConv_32487132627457
// MI455X (gfx1250) — compile-verified
//
#include <hip/hip_runtime.h>

typedef __attribute__((ext_vector_type(2))) float v2f;
typedef __attribute__((ext_vector_type(8))) float v8f;

#define GDIM 64

// --- 1. deg[i] = 1 (self-loop contribution) -------------------------------
__global__ void deg_init_kernel(float* __restrict__ deg, int n) {
  int i = blockIdx.x * blockDim.x + threadIdx.x;
  if (i < n) deg[i] = 1.0f;
}

// --- 2. deg[row[e]] += 1 ---------------------------------------------------
__global__ void deg_count_kernel(const int* __restrict__ row, float* __restrict__ deg, int e) {
  int i = blockIdx.x * blockDim.x + threadIdx.x;
  if (i < e) atomicAdd(&deg[row[i]], 1.0f);
}

// --- 3. dinv[i] = rsqrt(deg[i])  (deg >= 1 always) ------------------------
__global__ void dinv_kernel(const float* __restrict__ deg, float* __restrict__ dinv, int n) {
  int i = blockIdx.x * blockDim.x + threadIdx.x;
  if (i < n) dinv[i] = rsqrtf(deg[i]);
}

// --- 4. h = x @ W via V_WMMA_F32_16X16X4_F32 ------------------------------
// One wave per 16-row tile of x; 4 accumulators cover all 64 output columns.
// A (16x4 f32): lanes 0-15 hold {K,K+1} for M=lane; lanes 16-31 hold {K+2,K+3}.
// B (4x16 f32): lanes 0-15 hold rows {K,K+1} at N=lane; lanes 16-31 rows {K+2,K+3}.
// C/D (16x16 f32): VGPR v -> M = v (lanes 0-15, N=lane) / M = v+8 (lanes 16-31).
__global__ void gemm_wmma_kernel(const float* __restrict__ x, const float* __restrict__ w,
                                 float* __restrict__ h, int ntiles, int nrows) {
  const int gid  = blockIdx.x * blockDim.x + threadIdx.x;
  const int wave = gid >> 5;
  const int lane = threadIdx.x & 31;
  if (wave >= ntiles) return;              // wave-uniform: EXEC stays all-1s

  const int mbase = wave * 16;
  const int m     = lane & 15;             // A row / B,C column within tile
  const int kh    = (lane >> 4) << 1;      // 0 for lanes 0-15, 2 for lanes 16-31

  const float* xr = x + (size_t)(mbase + m) * GDIM;

  v8f c0 = {}, c1 = {}, c2 = {}, c3 = {};
  for (int k = 0; k < GDIM; k += 4) {
    v2f a;
    a.x = xr[k + kh];
    a.y = xr[k + kh + 1];

    const float* w0 = w + (size_t)(k + kh) * GDIM;
    const float* w1 = w0 + GDIM;
    v2f b0, b1, b2, b3;
    b0.x = w0[m];      b0.y = w1[m];
    b1.x = w0[16 + m]; b1.y = w1[16 + m];
    b2.x = w0[32 + m]; b2.y = w1[32 + m];
    b3.x = w0[48 + m]; b3.y = w1[48 + m];

    c0 = __builtin_amdgcn_wmma_f32_16x16x4_f32(false, a, false, b0, (short)0, c0, false, false);
    c1 = __builtin_amdgcn_wmma_f32_16x16x4_f32(false, a, false, b1, (short)0, c1, false, false);
    c2 = __builtin_amdgcn_wmma_f32_16x16x4_f32(false, a, false, b2, (short)0, c2, false, false);
    c3 = __builtin_amdgcn_wmma_f32_16x16x4_f32(false, a, false, b3, (short)0, c3, false, false);
  }

  const int ro = (lane >> 4) << 3;         // +8 rows for upper half-wave
#pragma unroll
  for (int v = 0; v < 8; ++v) {
    const int row = mbase + ro + v;
    if (row < nrows) {
      float* hr = h + (size_t)row * GDIM + m;
      hr[0]  = c0[v];
      hr[16] = c1[v];
      hr[32] = c2[v];
      hr[48] = c3[v];
    }
  }
}

// --- 5. out = dinv^2 * h  (self-loop term; also initializes out) ----------
__global__ void selfloop_kernel(const float* __restrict__ h, const float* __restrict__ dinv,
                                float* __restrict__ out, int total) {
  int t = blockIdx.x * blockDim.x + threadIdx.x;
  if (t < total) {
    const int i = t >> 6;                  // node = t / 64
    const float s = dinv[i];
    out[t] = s * s * h[t];
  }
}

// --- 6. out[col] += dinv[row]*dinv[col] * h[row]  (16 threads per edge) ---
__global__ void scatter_kernel(const int* __restrict__ row, const int* __restrict__ col,
                               const float* __restrict__ h, const float* __restrict__ dinv,
                               float* __restrict__ out, int e) {
  const long long t  = (long long)blockIdx.x * blockDim.x + threadIdx.x;
  const int ed       = (int)(t >> 4);
  if (ed >= e) return;
  const int chunk    = (int)(t & 15);      // which float4 of the 64-dim row

  const int r = row[ed];
  const int c = col[ed];
  const float coef = dinv[r] * dinv[c];

  const float4* hr = (const float4*)(h + (size_t)r * GDIM);
  const float4 v = hr[chunk];

  float* o = out + (size_t)c * GDIM + chunk * 4;
  atomicAdd(o + 0, coef * v.x);
  atomicAdd(o + 1, coef * v.y);
  atomicAdd(o + 2, coef * v.z);
  atomicAdd(o + 3, coef * v.w);
}

extern "C" void kernel_launch(void* const* d_in, const int* in_sizes, int n_in,
                              void* d_out, int out_size, void* d_ws, size_t ws_size,
                              hipStream_t stream) {
  const float* x  = (const float*)d_in[0];
  const float* w  = (const float*)d_in[1];
  const int*   ei = (const int*)d_in[2];
  float* out = (float*)d_out;

  const int n = in_sizes[0] / GDIM;        // 100000
  const int e = in_sizes[2] / 2;           // 1600000
  const int* row = ei;                     // edge_index[0]
  const int* col = ei + e;                 // edge_index[1]

  // workspace: deg[n] | dinv[n] | h[n*64]
  float* deg  = (float*)d_ws;
  float* dinv = deg + n;
  float* h    = dinv + n;

  const int B = 256;

  deg_init_kernel<<<(n + B - 1) / B, B, 0, stream>>>(deg, n);
  deg_count_kernel<<<(e + B - 1) / B, B, 0, stream>>>(row, deg, e);
  dinv_kernel<<<(n + B - 1) / B, B, 0, stream>>>(deg, dinv, n);

  const int ntiles = (n + 15) / 16;        // 6250 waves
  const long long gthreads = (long long)ntiles * 32;
  gemm_wmma_kernel<<<(int)((gthreads + B - 1) / B), B, 0, stream>>>(x, w, h, ntiles, n);

  const long long total = (long long)n * GDIM;
  selfloop_kernel<<<(int)((total + B - 1) / B), B, 0, stream>>>(h, dinv, out, (int)total);

  const long long st = (long long)e * 16;
  scatter_kernel<<<(int)((st + B - 1) / B), B, 0, stream>>>(row, col, h, dinv, out, e);
}